// GrCNetSpmm_7962869367666
// MI455X (gfx1250) — compile-verified
//
#include <hip/hip_runtime.h>
#include <stdint.h>

// ---------------------------------------------------------------------------
// Segment-sum scatter-add:  out[row[e], f] += edge_w[e, f]
//   E = 800000 edges, F = 64 features, N = 50000 rows  (f32)
// Memory-bound (AI ~0.23 FLOP/B, floor ~10us at 23.3 TB/s). CDNA5 plan:
//   * TDM (tensor_load_to_lds) DMAs contiguous 16KB edge_w tiles into LDS,
//     double-buffered, tracked with TENSORcnt (s_wait_tensorcnt).
//   * 8 waves/block consume tiles via ds_load_b128 and emit no-return
//     global_atomic_add_f32 into the 12.8MB output (L2-resident, 192MB L2).
//   * wave32: 16 lanes x float4 = one 256B edge row per half-wave.
// ---------------------------------------------------------------------------

typedef unsigned int u32x4 __attribute__((ext_vector_type(4)));
typedef int          i32x8 __attribute__((ext_vector_type(8)));
typedef int          i32x4 __attribute__((ext_vector_type(4)));

#define FEATS       64
#define TILE_EDGES  64
#define TILE_ELEMS  (TILE_EDGES * FEATS)   // 4096 f32
#define TILE_BYTES  (TILE_ELEMS * 4)       // 16 KB per buffer

// Issue one TDM descriptor: contiguous 1-D copy of TILE_ELEMS f32 from
// global `gsrc` into LDS byte offset `lds_off`. `valid_elems` bounds the
// tensor so a tail tile zero-fills instead of reading OOB.
__device__ __forceinline__ void tdm_issue(const float* gsrc, unsigned lds_off,
                                          unsigned valid_elems) {
    unsigned long long ga = (unsigned long long)(uintptr_t)gsrc;
    // Group 0: count=1 | lds_addr | global_addr[56:0] | type=2 ("image")
    u32x4 g0 = { 1u,
                 lds_off,
                 (unsigned)ga,
                 ((unsigned)(ga >> 32) & 0x01FFFFFFu) | (2u << 30) };
    // Group 1: wg_mask=0, data_size=2 (4B), no pad/iterate/barrier.
    //   tensor_dim0 = valid_elems (bits 79:48), tensor_dim1 = 1,
    //   tile_dim0 = TILE_ELEMS (bits 127:112), tile_dim1/2 = 0 (unused),
    //   tensor_dim0_stride = TILE_ELEMS.
    i32x8 g1 = { (int)0x00020000u,
                 (int)((valid_elems & 0xFFFFu) << 16),
                 (int)((valid_elems >> 16) | (1u << 16)),
                 (int)(TILE_ELEMS << 16),
                 0,
                 TILE_ELEMS,
                 0, 0 };
    // Groups 2/3: benign 1-element upper dims, zero tile dims.
    i32x4 g2 = { 1, 1, TILE_ELEMS, 0 };
    i32x4 g3 = { TILE_ELEMS, 1 << 16, 0, 0 };
    // clang-23 / therock-10.0 form: 6 args, extra int32x8 operand (zeroed).
    i32x8 gx = { 0, 0, 0, 0, 0, 0, 0, 0 };
    __builtin_amdgcn_tensor_load_to_lds(g0, g1, g2, g3, gx, 0);
}

__global__ void __launch_bounds__(256) zero4_kernel(float4* __restrict__ out, long n4) {
    long i = (long)blockIdx.x * blockDim.x + threadIdx.x;
    long stride = (long)gridDim.x * blockDim.x;
    float4 z = {0.f, 0.f, 0.f, 0.f};
    for (; i < n4; i += stride) out[i] = z;
}

__global__ void __launch_bounds__(256) seg_sum_tdm_kernel(
        const int*   __restrict__ rows,     // edge[0,:]  (E int32 entries)
        const float* __restrict__ edge_w,   // [E, 64] f32, row-major
        float*       __restrict__ out,      // [N, 64] f32 (pre-zeroed)
        long E, long ntiles) {
    __shared__ float4 lds[2][TILE_EDGES * 16];   // 2 x 16 KB double buffer
    const unsigned lds_base = (unsigned)(uintptr_t)(&lds[0][0]); // low32 = LDS offset

    const long first = blockIdx.x;
    const long step  = gridDim.x;
    if (first >= ntiles) return;

    const long total_elems = E * FEATS;

    // Prime the pipeline: wave 0 DMAs tile `first` into buffer 0.
    if (threadIdx.x == 0) {
        long rem = total_elems - first * (long)TILE_ELEMS;
        unsigned valid = (unsigned)(rem < TILE_ELEMS ? rem : TILE_ELEMS);
        tdm_issue(edge_w + first * (long)TILE_ELEMS, lds_base, valid);
    }

    int buf = 0;
    const int fq = threadIdx.x & 15;   // feature quad: floats [4*fq, 4*fq+4)
    const int e0 = threadIdx.x >> 4;   // local edge base 0..15

    for (long t = first; t < ntiles; t += step) {
        const long tn = t + step;
        if (threadIdx.x == 0) {
            if (tn < ntiles) {
                long rem = total_elems - tn * (long)TILE_ELEMS;
                unsigned valid = (unsigned)(rem < TILE_ELEMS ? rem : TILE_ELEMS);
                tdm_issue(edge_w + tn * (long)TILE_ELEMS,
                          lds_base + (unsigned)((buf ^ 1) * TILE_BYTES), valid);
                __builtin_amdgcn_s_wait_tensorcnt(1);  // tile t done (in-order)
            } else {
                __builtin_amdgcn_s_wait_tensorcnt(0);
            }
        }
        __syncthreads();   // publish tile t's LDS data to all 8 waves

        const float4* __restrict__ src = &lds[buf][0];
        #pragma unroll
        for (int i = 0; i < 4; ++i) {
            const int  el = e0 + i * 16;               // 0..63 within tile
            const long e  = t * (long)TILE_EDGES + el;
            if (e < E) {
                const int    row = rows[e];
                const float4 v   = src[el * 16 + fq];  // ds_load_b128
                float* dst = out + (long)row * FEATS + fq * 4;
                // no-return f32 atomics -> STOREcnt, resolve in L2
                __hip_atomic_fetch_add(dst + 0, v.x, __ATOMIC_RELAXED, __HIP_MEMORY_SCOPE_AGENT);
                __hip_atomic_fetch_add(dst + 1, v.y, __ATOMIC_RELAXED, __HIP_MEMORY_SCOPE_AGENT);
                __hip_atomic_fetch_add(dst + 2, v.z, __ATOMIC_RELAXED, __HIP_MEMORY_SCOPE_AGENT);
                __hip_atomic_fetch_add(dst + 3, v.w, __ATOMIC_RELAXED, __HIP_MEMORY_SCOPE_AGENT);
            }
        }
        __syncthreads();   // all waves done with buf before TDM may overwrite it
        buf ^= 1;
    }
}

extern "C" void kernel_launch(void* const* d_in, const int* in_sizes, int n_in,
                              void* d_out, int out_size, void* d_ws, size_t ws_size,
                              hipStream_t stream) {
    const int*   edge   = (const int*)d_in[0];    // [2, E]; rows = edge[0:E]
    const float* edge_w = (const float*)d_in[1];  // [E, 64]
    float*       out    = (float*)d_out;          // [N, 64]

    const long E      = (long)in_sizes[0] / 2;
    const long ntiles = (E + TILE_EDGES - 1) / TILE_EDGES;

    // Harness poisons d_out; segment-sum needs a zeroed accumulator each call.
    const long n4 = (long)out_size / 4;
    zero4_kernel<<<512, 256, 0, stream>>>((float4*)out, n4);

    long g = ntiles < 2048 ? ntiles : 2048;
    seg_sum_tdm_kernel<<<(int)g, 256, 0, stream>>>(edge, edge_w, out, E, ntiles);
}